// PANPooling_56908316672641
// MI455X (gfx1250) — compile-verified
//
#include <hip/hip_runtime.h>
#include <math.h>

// ---------------------------------------------------------------------------
// PANPooling for MI455X (gfx1250, wave32).
// Pipeline: WMMA GEMV -> atomic scatter-add -> tanh/key build -> bitonic
// top-K sort (fits in 192MB L2) -> gather/scale x_out -> edge lexsort
// (bitonic, 33.5MB keys, L2-resident) -> keep flags -> LDS two-level scan ->
// stable compaction. Streaming outputs use non-temporal stores to keep L2
// for the sort keys.
// ---------------------------------------------------------------------------

typedef __attribute__((ext_vector_type(2))) float v2f;
typedef __attribute__((ext_vector_type(8))) float v8f;

#define PP_N     100000
#define PP_D     256
#define PP_E     3200000
#define PP_K     50000
#define PP_NPAD  (1 << 17)       // 131072 >= N
#define PP_EPAD  (1 << 22)       // 4194304 >= E
#define PP_NB    3125            // E / 1024 (exact)

// ---------------- init node scratch (score2 accumulator, mask) -------------
__global__ void pp_init_node(float* __restrict__ s2, int* __restrict__ mask) {
  int i = blockIdx.x * blockDim.x + threadIdx.x;
  if (i < PP_N) { s2[i] = 0.0f; mask[i] = -1; }
}

// ---------------- score1 = x @ p via V_WMMA_F32_16X16X4_F32 ----------------
// One wave32 computes 16 row-dot-products. A = 16x4 tile of x rows
// (lane m<16 holds K={0,1}, lanes 16..31 hold K={2,3}); B has p[k0..k0+3]
// in column N=0 only, so D column 0 accumulates dot(x_row, p).
// p chunk is loaded wave-uniformly (scalar loads) and selected into the B
// fragment with branch-free cndmask — no EXEC manipulation in the hot loop.
__global__ void pp_score1_wmma(const float* __restrict__ x,
                               const float* __restrict__ p,
                               float* __restrict__ s1) {
  const int lane  = threadIdx.x & 31;
  const int wave  = (blockIdx.x * blockDim.x + threadIdx.x) >> 5;
  const int row0  = wave * 16;
  if (row0 >= PP_N) return;                 // wave-uniform exit
  const int m     = lane & 15;
  const bool hi   = (lane >> 4) != 0;       // lanes 16..31 carry K = {2,3}
  const int khalf = hi ? 2 : 0;
  const bool isn0 = (m == 0);               // only N=0 column of B is nonzero
  int row = row0 + m;
  if (row >= PP_N) row = PP_N - 1;          // safety (N % 16 == 0, never taken)
  const float* __restrict__ xr = x + (size_t)row * PP_D;
  v8f acc = {};
  for (int k0 = 0; k0 < PP_D; k0 += 4) {
    __builtin_prefetch(xr + k0 + 64, 0, 3); // near-cache stream-ahead hint
    v2f a, b;
    a.x = xr[k0 + khalf];
    a.y = xr[k0 + khalf + 1];
    // wave-uniform p chunk -> constant-cache scalar loads
    float p0 = p[k0 + 0], p1 = p[k0 + 1], p2 = p[k0 + 2], p3 = p[k0 + 3];
    float bx = hi ? p2 : p0;
    float by = hi ? p3 : p1;
    b.x = isn0 ? bx : 0.0f;                 // v_cndmask, no branch
    b.y = isn0 ? by : 0.0f;
    acc = __builtin_amdgcn_wmma_f32_16x16x4_f32(false, a, false, b,
                                                (short)0, acc, false, false);
  }
  // D column N=0: lane 0 holds M=0..7 in acc[0..7], lane 16 holds M=8..15.
  if (isn0) {
    int base = row0 + (hi ? 8 : 0);
#pragma unroll
    for (int r = 0; r < 8; ++r)
      if (base + r < PP_N) s1[base + r] = acc[r];
  }
}

// ---------------- score2[col] += edge_weight (order-free sum) --------------
__global__ void pp_scatter_s2(const int* __restrict__ col,
                              const float* __restrict__ ea,
                              float* __restrict__ s2) {
  int e = blockIdx.x * blockDim.x + threadIdx.x;
  if (e < PP_E) atomicAdd(&s2[col[e]], ea[e]);
}

// ------- score = tanh(b0*s1 + b1*s2); build descending-sort keys -----------
__global__ void pp_combine_keys(const float* __restrict__ s1,
                                const float* __restrict__ s2,
                                const float* __restrict__ beta,
                                float* __restrict__ score,
                                unsigned long long* __restrict__ keys) {
  int i = blockIdx.x * blockDim.x + threadIdx.x;
  if (i >= PP_NPAD) return;
  if (i < PP_N) {
    float s = tanhf(beta[0] * s1[i] + beta[1] * s2[i]);
    score[i] = s;
    unsigned u = __float_as_uint(s);
    u ^= (u >> 31) ? 0xFFFFFFFFu : 0x80000000u;  // float -> ascending uint
    u = ~u;                                      // descending
    keys[i] = ((unsigned long long)u << 32) | (unsigned)i;
  } else {
    keys[i] = 0xFFFFFFFFFFFFFFFFull;             // pad sorts last
  }
}

// ---------------- generic bitonic compare-exchange pass --------------------
__global__ void pp_bitonic_pass(unsigned long long* __restrict__ keys,
                                int j, int k, int n) {
  int i = blockIdx.x * blockDim.x + threadIdx.x;
  if (i >= n) return;
  int ixj = i ^ j;
  if (ixj > i) {
    unsigned long long a = keys[i], b = keys[ixj];
    bool asc = ((i & k) == 0);
    if ((a > b) == asc) { keys[i] = b; keys[ixj] = a; }
  }
}

// ------- top-K: x_out = x[perm]*topv, batch_out, mask[perm] = rank ---------
__global__ void pp_write_topk(const unsigned long long* __restrict__ keys,
                              const float* __restrict__ score,
                              const float* __restrict__ x,
                              const int* __restrict__ batch,
                              float* __restrict__ x_out,
                              int* __restrict__ batch_out,
                              int* __restrict__ mask) {
  int i   = blockIdx.x;                       // 0..K-1 (rank)
  int idx = (int)(keys[i] & 0xFFFFFFFFull);
  float tv = score[idx];
  int t = threadIdx.x;                        // 0..255 == D
  // write-once stream: non-temporal store keeps L2 for the sort keys
  __builtin_nontemporal_store(x[(size_t)idx * PP_D + t] * tv,
                              &x_out[(size_t)i * PP_D + t]);
  if (t == 0) { batch_out[i] = batch[idx]; mask[idx] = i; }
}

// ---------------- edge lexsort keys: (row << 39) | (col << 22) | e ---------
__global__ void pp_build_edge_keys(const int* __restrict__ row,
                                   const int* __restrict__ col,
                                   unsigned long long* __restrict__ ekeys) {
  int e = blockIdx.x * blockDim.x + threadIdx.x;
  if (e >= PP_EPAD) return;
  if (e < PP_E) {
    ekeys[e] = ((unsigned long long)(unsigned)row[e] << 39) |
               ((unsigned long long)(unsigned)col[e] << 22) |
               (unsigned long long)(unsigned)e;
  } else {
    ekeys[e] = 0xFFFFFFFFFFFFFFFFull;
  }
}

// ---------------- keep flags over sorted edges -----------------------------
__global__ void pp_keep_flags(const unsigned long long* __restrict__ ekeys,
                              const int* __restrict__ mask,
                              unsigned* __restrict__ flags) {
  int s = blockIdx.x * blockDim.x + threadIdx.x;
  if (s >= PP_E) return;
  unsigned long long key = ekeys[s];
  int col = (int)((key >> 22) & 0x1FFFFull);
  int row = (int)(key >> 39);
  int r = mask[col];                // r = mask[ei_t[0]] = mask[col]
  int c = mask[row];                // c = mask[ei_t[1]] = mask[row]
  flags[s] = (r >= 0 && c >= 0) ? 1u : 0u;
}

// ---------------- two-level exclusive scan (LDS) ---------------------------
__global__ void pp_block_scan(const unsigned* __restrict__ flags,
                              unsigned* __restrict__ pos,
                              unsigned* __restrict__ bsum) {
  __shared__ unsigned sh[256];
  int b = blockIdx.x, t = threadIdx.x;
  int base = b * 1024 + t * 4;
  unsigned v[4]; unsigned local = 0;
#pragma unroll
  for (int r = 0; r < 4; ++r) {
    int i = base + r;
    v[r] = (i < PP_E) ? flags[i] : 0u;
    local += v[r];
  }
  sh[t] = local;
  __syncthreads();
  for (int off = 1; off < 256; off <<= 1) {
    unsigned add = (t >= off) ? sh[t - off] : 0u;
    __syncthreads();
    sh[t] += add;
    __syncthreads();
  }
  unsigned run = (t == 0) ? 0u : sh[t - 1];
#pragma unroll
  for (int r = 0; r < 4; ++r) {
    int i = base + r;
    if (i < PP_E) pos[i] = run;
    run += v[r];
  }
  if (t == 255) bsum[b] = sh[255];
}

__global__ void pp_scan_bsums(const unsigned* __restrict__ bsum,
                              unsigned* __restrict__ boff,
                              int* __restrict__ num_kept_out) {
  __shared__ unsigned sh[256];
  __shared__ unsigned carry;
  int t = threadIdx.x;
  if (t == 0) carry = 0;
  __syncthreads();
  for (int base = 0; base < PP_NB; base += 256) {
    int i = base + t;
    sh[t] = (i < PP_NB) ? bsum[i] : 0u;
    __syncthreads();
    for (int off = 1; off < 256; off <<= 1) {
      unsigned add = (t >= off) ? sh[t - off] : 0u;
      __syncthreads();
      sh[t] += add;
      __syncthreads();
    }
    if (i < PP_NB) boff[i] = carry + ((t == 0) ? 0u : sh[t - 1]);
    __syncthreads();
    if (t == 0) carry += sh[255];
    __syncthreads();
  }
  if (t == 0) *num_kept_out = (int)carry;
}

// ---------------- initialize edge outputs (-1 / 0 padding) -----------------
__global__ void pp_init_edge_out(int* __restrict__ ei_out,
                                 float* __restrict__ ea_out) {
  int j = blockIdx.x * blockDim.x + threadIdx.x;
  if (j < 2 * PP_E) __builtin_nontemporal_store(-1, &ei_out[j]);
  if (j < PP_E)     __builtin_nontemporal_store(0.0f, &ea_out[j]);
}

// ---------------- stable compaction of kept edges --------------------------
__global__ void pp_scatter_kept(const unsigned long long* __restrict__ ekeys,
                                const int* __restrict__ mask,
                                const unsigned* __restrict__ flags,
                                const unsigned* __restrict__ pos,
                                const unsigned* __restrict__ boff,
                                const float* __restrict__ ea_in,
                                int* __restrict__ ei_out,
                                float* __restrict__ ea_out) {
  int s = blockIdx.x * blockDim.x + threadIdx.x;
  if (s >= PP_E || !flags[s]) return;
  unsigned long long key = ekeys[s];
  int e   = (int)(key & 0x3FFFFFull);
  int col = (int)((key >> 22) & 0x1FFFFull);
  int row = (int)(key >> 39);
  unsigned j = pos[s] + boff[s >> 10];
  ei_out[j]        = mask[col];   // r
  ei_out[PP_E + j] = mask[row];   // c
  ea_out[j]        = ea_in[e];
}

// ---------------------------------------------------------------------------
extern "C" void kernel_launch(void* const* d_in, const int* in_sizes, int n_in,
                              void* d_out, int out_size, void* d_ws, size_t ws_size,
                              hipStream_t stream) {
  (void)in_sizes; (void)n_in; (void)out_size;
  const float* x    = (const float*)d_in[0];
  const int*   ei   = (const int*)d_in[1];      // [2,E]: row = ei, col = ei+E
  const float* ea   = (const float*)d_in[2];    // [E,1]
  const int*   bat  = (const int*)d_in[3];
  const float* p    = (const float*)d_in[4];
  const float* beta = (const float*)d_in[5];
  const int* rowi = ei;
  const int* coli = ei + PP_E;

  // ---- output regions (concatenated flat, ints bit-cast into the buffer) --
  char* ob = (char*)d_out;
  float* x_out   = (float*)ob;                                           // K*D
  int*   ei_out  = (int*)  (ob + (size_t)PP_K * PP_D * 4);               // 2E
  float* ea_out  = (float*)(ob + ((size_t)PP_K * PP_D + 2ull*PP_E) * 4); // E
  int*   b_out   = (int*)  (ob + ((size_t)PP_K * PP_D + 3ull*PP_E) * 4); // K
  int*   nk_out  = (int*)  (ob + ((size_t)PP_K * PP_D + 3ull*PP_E + PP_K) * 4);

  // ---- workspace layout (4-byte aligned offsets; key arrays 8-byte aligned) --------------------------
  char* ws = (char*)d_ws;
  float* s1   = (float*)(ws + 0);                        //  400000 B
  float* s2   = (float*)(ws + 400000);                   //  400000 B
  float* sc   = (float*)(ws + 800000);                   //  400000 B
  int*   mask = (int*)  (ws + 1200000);                  //  400000 B
  unsigned long long* nkeys = (unsigned long long*)(ws + 1600000);  // 1 MiB
  unsigned long long* ekeys = (unsigned long long*)(ws + 2648576);  // 32 MiB
  unsigned* flags = (unsigned*)(ws + 36203008);          // 12.8 MB
  unsigned* pos   = (unsigned*)(ws + 49003008);          // 12.8 MB
  unsigned* bsum  = (unsigned*)(ws + 61803008);          // 12.5 KB
  unsigned* boff  = (unsigned*)(ws + 61815508);          // 12.5 KB
  if (ws_size < 61828008ull) return;

  // 1) init node scratch
  pp_init_node<<<(PP_N + 255) / 256, 256, 0, stream>>>(s2, mask);
  // 2) GEMV via WMMA: 6250 waves of 16 rows, 8 waves/block
  pp_score1_wmma<<<(PP_N / 16 + 7) / 8, 256, 0, stream>>>(x, p, s1);
  // 3) scatter-add edge weights over col
  pp_scatter_s2<<<(PP_E + 255) / 256, 256, 0, stream>>>(coli, ea, s2);
  // 4) combine + build node sort keys
  pp_combine_keys<<<PP_NPAD / 256, 256, 0, stream>>>(s1, s2, beta, sc, nkeys);
  // 5) bitonic sort node keys (1 MB, L2-resident)
  for (int k = 2; k <= PP_NPAD; k <<= 1)
    for (int j = k >> 1; j > 0; j >>= 1)
      pp_bitonic_pass<<<PP_NPAD / 256, 256, 0, stream>>>(nkeys, j, k, PP_NPAD);
  // 6) emit top-K rows, batch, and rank mask
  pp_write_topk<<<PP_K, PP_D, 0, stream>>>(nkeys, sc, x, bat, x_out, b_out, mask);
  // 7) edge lexsort keys
  pp_build_edge_keys<<<PP_EPAD / 256, 256, 0, stream>>>(rowi, coli, ekeys);
  // 8) bitonic sort edge keys (33.5 MB, fits 192 MB L2)
  for (int k = 2; k <= PP_EPAD; k <<= 1)
    for (int j = k >> 1; j > 0; j >>= 1)
      pp_bitonic_pass<<<PP_EPAD / 256, 256, 0, stream>>>(ekeys, j, k, PP_EPAD);
  // 9) keep flags over sorted edges
  pp_keep_flags<<<(PP_E + 255) / 256, 256, 0, stream>>>(ekeys, mask, flags);
  // 10) two-level exclusive scan
  pp_block_scan<<<PP_NB, 256, 0, stream>>>(flags, pos, bsum);
  pp_scan_bsums<<<1, 256, 0, stream>>>(bsum, boff, nk_out);
  // 11) pad outputs then stable-compact kept edges
  pp_init_edge_out<<<(2 * PP_E + 255) / 256, 256, 0, stream>>>(ei_out, ea_out);
  pp_scatter_kept<<<(PP_E + 255) / 256, 256, 0, stream>>>(ekeys, mask, flags,
                                                          pos, boff, ea,
                                                          ei_out, ea_out);
}